// VectorQuantizerEMA_10668698763656
// MI455X (gfx1250) — compile-verified
//
#include <hip/hip_runtime.h>
#include <hip/hip_bf16.h>
#include <math.h>

// ---------------- problem constants ----------------
#define BATCH 32
#define CH    256           // C == D
#define HW    1024          // H*W
#define NVEC  32768         // B*H*W
#define KCODE 1024
#define DDIM  256
#define OUT_ELEMS   8388608     // B*C*H*W
#define ENC_OFFSET  8388610     // after out + loss + entropy
#define LOSS_OFFSET 8388608
#define ENT_OFFSET  8388609

typedef __attribute__((ext_vector_type(2))) float v2f;
typedef __attribute__((ext_vector_type(8))) float v8f;

// ---------------- 1) BCHW -> [N, D] transpose ----------------
__global__ __launch_bounds__(256) void vq_transpose_kernel(
    const float* __restrict__ in, float* __restrict__ xflat) {
  __shared__ float tile[32][33];
  const int bid = blockIdx.x;           // 8192 blocks
  const int b   = bid >> 8;             // / 256
  const int r   = bid & 255;
  const int ct  = r >> 5;               // 8 c-tiles
  const int wt  = r & 31;               // 32 hw-tiles
  const int tx  = threadIdx.x & 31;
  const int ty  = threadIdx.x >> 5;     // 0..7
#pragma unroll
  for (int j = 0; j < 32; j += 8) {
    const int c  = ct * 32 + ty + j;
    const int hw = wt * 32 + tx;
    tile[ty + j][tx] = in[(b * CH + c) * HW + hw];
  }
  __syncthreads();
#pragma unroll
  for (int j = 0; j < 32; j += 8) {
    const int hw = wt * 32 + ty + j;
    const int c  = ct * 32 + tx;
    xflat[(b * HW + hw) * DDIM + c] = tile[tx][ty + j];
  }
}

// ---------------- 2) wnorm[k] = ||w_k||^2 ----------------
__global__ __launch_bounds__(256) void vq_wnorm_kernel(
    const float* __restrict__ weight, float* __restrict__ wnorm) {
  const int k = blockIdx.x * 256 + threadIdx.x;  // 4 blocks
  const float4* row = (const float4*)(weight + k * DDIM);
  float s = 0.f;
#pragma unroll
  for (int q = 0; q < 64; ++q) {
    float4 v = row[q];
    s += v.x * v.x + v.y * v.y + v.z * v.z + v.w * v.w;
  }
  wnorm[k] = s;
}

// ---------------- 3) WMMA distance GEMM + argmin ----------------
// 256 blocks x 256 threads (8 waves). Wave owns 16 rows (A tile in regs).
// Per 16-code column tile: stage w^T pair-major in LDS, 64x v_wmma_f32_16x16x4_f32.
__global__ __launch_bounds__(256) void vq_argmin_kernel(
    const float* __restrict__ xflat, const float* __restrict__ weight,
    const float* __restrict__ wnorm, int* __restrict__ idx,
    int* __restrict__ counts) {
  __shared__ v2f ldsB[128 * 16];  // [pair p = d/2][j] -> (w[k0+j][2p], w[k0+j][2p+1])

  const int lane  = threadIdx.x & 31;
  const int wave  = threadIdx.x >> 5;
  const int jlane = lane & 15;    // M (for A) / N (for B,C)
  const int half  = lane >> 4;    // K-pair select
  const int rowbase = blockIdx.x * 128 + wave * 16;

  // A fragments for all 64 K-chunks, resident in VGPRs (128 regs/lane).
  // A layout (16x4 f32): lane holds row = jlane, d = 4*dc + 2*half + {0,1}.
  v2f a[64];
  {
    const float* ap = xflat + (rowbase + jlane) * DDIM + 2 * half;
#pragma unroll
    for (int dc = 0; dc < 64; ++dc)
      a[dc] = *(const v2f*)(ap + 4 * dc);
  }

  float minv[8];
  int   minc[8];
#pragma unroll
  for (int i = 0; i < 8; ++i) { minv[i] = 3.4e38f; minc[i] = 0; }

  for (int kt = 0; kt < 64; ++kt) {
    const int k0 = kt * 16;
    __syncthreads();
    {  // stage B tile: 16 codes x 256 d, transposed pair-major
      const int j    = threadIdx.x >> 4;   // code 0..15
      const int pblk = threadIdx.x & 15;   // covers pairs pblk*8 .. +7
      const float4* wrow =
          (const float4*)(weight + (k0 + j) * DDIM + pblk * 16);
#pragma unroll
      for (int q = 0; q < 4; ++q) {
        float4 v = wrow[q];
        v2f lo; lo.x = v.x; lo.y = v.y;
        v2f hi; hi.x = v.z; hi.y = v.w;
        ldsB[(pblk * 8 + q * 2 + 0) * 16 + j] = lo;
        ldsB[(pblk * 8 + q * 2 + 1) * 16 + j] = hi;
      }
    }
    __syncthreads();

    const float wn = wnorm[k0 + jlane];
    v8f acc = {};
#pragma unroll
    for (int dc = 0; dc < 64; ++dc) {
      // B layout (4x16 f32): lane holds col = jlane, d = 4*dc + 2*half + {0,1}
      v2f bfrag = ldsB[(2 * dc + half) * 16 + jlane];
      acc = __builtin_amdgcn_wmma_f32_16x16x4_f32(
          false, a[dc], false, bfrag, (short)0, acc, false, false);
    }
    // dist-proxy = ||w||^2 - 2 x.w  (||x||^2 is argmin-invariant)
#pragma unroll
    for (int i = 0; i < 8; ++i) {
      float v = wn - 2.0f * acc[i];
      if (v < minv[i]) { minv[i] = v; minc[i] = k0 + jlane; }
    }
  }

  // C/D layout: VGPR i, lanes 0-15 -> M=i ; lanes 16-31 -> M=8+i.
  // Reduce min over the 16 lanes of each half (first-index tie-break).
#pragma unroll
  for (int i = 0; i < 8; ++i) {
    float v = minv[i];
    int   c = minc[i];
#pragma unroll
    for (int off = 8; off >= 1; off >>= 1) {
      float ov = __shfl_xor(v, off, 32);
      int   oc = __shfl_xor(c, off, 32);
      if (ov < v || (ov == v && oc < c)) { v = ov; c = oc; }
    }
    if (jlane == 0) {
      const int row = rowbase + i + 8 * half;
      idx[row] = c;
      atomicAdd(&counts[c], 1);
    }
  }
}

// ---------------- 4) one-hot encodings [N, K] ----------------
__global__ __launch_bounds__(256) void vq_encodings_kernel(
    const int* __restrict__ idx, float* __restrict__ enc) {
  const int n  = blockIdx.x;
  const int t  = threadIdx.x;
  const int id = idx[n];
  v2f* row = (v2f*)(enc + (size_t)n * KCODE);  // 8B-aligned slice of d_out
#pragma unroll
  for (int s = 0; s < 512; s += 256) {
    const int col2 = s + t;           // covers cols 2*col2, 2*col2+1
    v2f v; v.x = 0.f; v.y = 0.f;
    if ((id >> 1) == col2) { if (id & 1) v.y = 1.f; else v.x = 1.f; }
    row[col2] = v;
  }
}

// ---------------- 5) quantized output (BCHW gather) ----------------
__global__ __launch_bounds__(256) void vq_gather_kernel(
    const float* __restrict__ weight, const int* __restrict__ idx,
    float* __restrict__ out) {
  const int e  = blockIdx.x * 256 + threadIdx.x;  // 32768 blocks
  const int hw = e & (HW - 1);
  const int c  = (e >> 10) & (CH - 1);
  const int b  = e >> 18;
  const int id = idx[(b << 10) + hw];
  out[e] = weight[id * DDIM + c];   // weight is L2-resident (1 MB)
}

// ---------------- 6) commitment loss sum ----------------
__global__ __launch_bounds__(256) void vq_loss_kernel(
    const float* __restrict__ xflat, const float* __restrict__ weight,
    const int* __restrict__ idx, double* __restrict__ lsum) {
  __shared__ float red[256];
  const int n = blockIdx.x;        // one row per block
  const int d = threadIdx.x;
  const int id = idx[n];
  const float diff = weight[id * DDIM + d] - xflat[n * DDIM + d];
  red[d] = diff * diff;
  __syncthreads();
#pragma unroll
  for (int s = 128; s > 0; s >>= 1) {
    if (d < s) red[d] += red[d + s];
    __syncthreads();
  }
  if (d == 0) atomicAdd(lsum, (double)red[0]);
}

// ---------------- 7) finalize loss + entropy ----------------
__global__ __launch_bounds__(256) void vq_finalize_kernel(
    const int* __restrict__ counts, const double* __restrict__ lsum,
    float* __restrict__ out_loss, float* __restrict__ out_ent) {
  __shared__ float red[256];
  const int t = threadIdx.x;
  float e = 0.f;
#pragma unroll
  for (int k = t; k < KCODE; k += 256) {
    const float p = (float)counts[k] * (1.0f / (float)NVEC);
    e += -p * logf(p + 1e-10f);
  }
  red[t] = e;
  __syncthreads();
#pragma unroll
  for (int s = 128; s > 0; s >>= 1) {
    if (t < s) red[t] += red[t + s];
    __syncthreads();
  }
  if (t == 0) {
    *out_ent  = red[0];
    *out_loss = 0.25f * (float)(*lsum / (double)OUT_ELEMS);
  }
}

extern "C" void kernel_launch(void* const* d_in, const int* in_sizes, int n_in,
                              void* d_out, int out_size, void* d_ws, size_t ws_size,
                              hipStream_t stream) {
  const float* inputs = (const float*)d_in[0];   // [32,256,32,32] f32
  const float* weight = (const float*)d_in[1];   // [1024,256] f32
  float* out      = (float*)d_out;               // 8388608
  float* out_loss = out + LOSS_OFFSET;
  float* out_ent  = out + ENT_OFFSET;
  float* enc      = out + ENC_OFFSET;            // 33554432 one-hot floats

  // workspace layout
  char*   ws     = (char*)d_ws;
  float*  xflat  = (float*)ws;                          // 33,554,432 B
  float*  wnorm  = (float*)(ws + 33554432);             // 4 KB
  int*    idx    = (int*)(ws + 33554432 + 4096);        // 128 KB
  int*    counts = (int*)(ws + 33554432 + 4096 + 131072);       // 4 KB
  double* lsum   = (double*)(ws + 33554432 + 4096 + 131072 + 4096);

  // zero counts + loss accumulator (contiguous region)
  hipMemsetAsync(counts, 0, 4096 + 16, stream);

  vq_transpose_kernel<<<8192, 256, 0, stream>>>(inputs, xflat);
  vq_wnorm_kernel<<<4, 256, 0, stream>>>(weight, wnorm);
  vq_argmin_kernel<<<256, 256, 0, stream>>>(xflat, weight, wnorm, idx, counts);
  vq_encodings_kernel<<<NVEC, 256, 0, stream>>>(idx, enc);
  vq_gather_kernel<<<OUT_ELEMS / 256, 256, 0, stream>>>(weight, idx, out);
  vq_loss_kernel<<<NVEC, 256, 0, stream>>>(xflat, weight, idx, lsum);
  vq_finalize_kernel<<<1, 256, 0, stream>>>(counts, lsum, out_loss, out_ent);
}